// CrossWinAttention_15564961481522
// MI455X (gfx1250) — compile-verified
//
#include <hip/hip_runtime.h>
#include <math.h>

// ---------------------------------------------------------------------------
// CrossWinAttention for MI455X (gfx1250), full fp32 via V_WMMA_F32_16X16X4_F32
//   b=4, n=4, x=y=8, w=8, d=128, H=4, Dh=32, L=64, Q=K=256
//   keep_q = 192 (stable top-k on query saliency), topk = 64 keys per query
// ---------------------------------------------------------------------------

typedef float v2f __attribute__((ext_vector_type(2)));
typedef float v8f __attribute__((ext_vector_type(8)));

#define NB      4        // batch
#define NWIN    4        // n
#define LL      64       // x*y
#define QQ      256      // n*w1*w2
#define DHEAD   32
#define NH      4
#define HD      128
#define DD      128
#define BHTOT   16       // NB*NH
#define KEEPQ   192
#define TOPKK   64
#define ROWSTR  (QQ*DHEAD)   // 8192 floats per (bh,l) slab
#define QSCALE  0.17677669529663687f   // 1/sqrt(32)

// fp32 WMMA 16x16x4 accumulate step:  acc += A(16x4) * B(4x16)
//   A frag (v2f): lane L, vgpr j -> M = L%16, K = j + 2*(L/16)
//   B frag (v2f): lane L, vgpr j -> N = L%16, K = j + 2*(L/16)
//   C/D  (v8f):   lane L, vgpr r -> M = r + 8*(L/16), N = L%16
__device__ __forceinline__ v8f wmma_f32(v2f a, v2f b, v8f c) {
  return __builtin_amdgcn_wmma_f32_16x16x4_f32(false, a, false, b, (short)0, c,
                                               false, false);
}

// ---------------------------------------------------------------------------
// Kernel 1: LayerNorm + 128x128 projection, writing head-major (BH,L,Q,Dh)
//   rows ordered (b, l, r);  r = (n, w1, w2);  input is (b,n,x,y,w1,w2,d)
// ---------------------------------------------------------------------------
__global__ __launch_bounds__(256)
void ln_proj_kernel(const float* __restrict__ X, const float* __restrict__ g,
                    const float* __restrict__ be, const float* __restrict__ W,
                    const float* __restrict__ bias, float* __restrict__ out) {
  __shared__ float As[16 * 128];
  __shared__ float redS[16 * 16];
  __shared__ float redS2[16 * 16];
  __shared__ float rowMean[16], rowRstd[16];

  const int t = threadIdx.x;
  const int m = t >> 4;      // row within 16-row tile
  const int s16 = t & 15;
  const int rowbase = blockIdx.x * 16;

  // ---- stage raw row + LN statistics ----
  {
    int grow = rowbase + m;
    int b = grow >> 14;              // 64*256 rows per batch
    int l = (grow >> 8) & 63;
    int r = grow & 255;
    int n = r >> 6;
    int w12 = r & 63;
    const float* xrow = X + ((((size_t)(b * 4 + n)) * 64 + l) * 64 + w12) * 128;
    float s = 0.f, s2 = 0.f;
#pragma unroll
    for (int i = 0; i < 8; ++i) {
      int c = s16 + 16 * i;
      float x = xrow[c];
      As[m * 128 + c] = x;
      s += x;
      s2 += x * x;
    }
    redS[m * 16 + s16] = s;
    redS2[m * 16 + s16] = s2;
  }
  __syncthreads();
  if (s16 == 0) {
    float ts = 0.f, ts2 = 0.f;
    for (int i = 0; i < 16; ++i) { ts += redS[m * 16 + i]; ts2 += redS2[m * 16 + i]; }
    float mean = ts * (1.f / 128.f);
    float var = ts2 * (1.f / 128.f) - mean * mean;  // biased variance (torch LN)
    rowMean[m] = mean;
    rowRstd[m] = rsqrtf(var + 1e-5f);
  }
  __syncthreads();
  {
    float mean = rowMean[m], rstd = rowRstd[m];
#pragma unroll
    for (int i = 0; i < 8; ++i) {
      int c = s16 + 16 * i;
      As[m * 128 + c] = (As[m * 128 + c] - mean) * rstd * g[c] + be[c];
    }
  }
  __syncthreads();

  // ---- GEMM: 8 waves, one 16-col tile each; K = 128 ----
  const int wave = t >> 5;
  const int lane = t & 31;
  const int hi = lane >> 4;
  const int ln16 = lane & 15;
  const int tileN = wave * 16;

  v8f acc = {0.f, 0.f, 0.f, 0.f, 0.f, 0.f, 0.f, 0.f};
#pragma unroll 4
  for (int k0 = 0; k0 < 128; k0 += 4) {
    v2f a, b;
    a.x = As[ln16 * 128 + k0 + 2 * hi];
    a.y = As[ln16 * 128 + k0 + 1 + 2 * hi];
    b.x = W[(size_t)(k0 + 2 * hi) * 128 + tileN + ln16];
    b.y = W[(size_t)(k0 + 1 + 2 * hi) * 128 + tileN + ln16];
    acc = wmma_f32(a, b, acc);
  }
  const int col = tileN + ln16;
  const float bv = bias[col];
  const int h = col >> 5, dh = col & 31;
#pragma unroll
  for (int r = 0; r < 8; ++r) {
    int M = r + 8 * hi;
    int gr = rowbase + M;
    int b2 = gr >> 14;
    int l2 = (gr >> 8) & 63;
    int r2 = gr & 255;
    out[(((size_t)(b2 * 4 + h)) * 64 + l2) * ROWSTR + r2 * 32 + dh] = acc[r] + bv;
  }
}

// ---------------------------------------------------------------------------
// Kernel 2: query saliency pruning (stable top-192 of 256) + pre-scale by
// 1/sqrt(Dh).  One block per (bh,l) slab, one thread per query.
// ---------------------------------------------------------------------------
__global__ __launch_bounds__(256)
void prune_q_kernel(float* __restrict__ qbuf) {
  __shared__ float sal[256];
  const int q = threadIdx.x;
  float* base = qbuf + (size_t)blockIdx.x * ROWSTR;
  float v[32];
  float sq = 0.f;
#pragma unroll
  for (int i = 0; i < 32; ++i) {
    v[i] = base[q * 32 + i];
    sq += v[i] * v[i];
  }
  sal[q] = sq;
  __syncthreads();
  int rank = 0;
  for (int j = 0; j < 256; ++j) {
    float o = sal[j];
    rank += (o > sq) || (o == sq && j < q);  // stable: earlier index wins ties
  }
  const float sc = (rank < KEEPQ) ? QSCALE : 0.f;
#pragma unroll
  for (int i = 0; i < 32; ++i) base[q * 32 + i] = v[i] * sc;
}

// ---------------------------------------------------------------------------
// Kernel 3: attention for one (bh, l, 16-query tile):
//   logits (WMMA) -> exact stable top-64 via binary search on sortable bits
//   -> softmax -> att @ V (WMMA)
// ---------------------------------------------------------------------------
__global__ __launch_bounds__(256)
void attn_kernel(const float* __restrict__ qp, const float* __restrict__ kb,
                 const float* __restrict__ vb, float* __restrict__ ab) {
  __shared__ float At[16 * 32];
  __shared__ float S[16 * 256];

  const int t = threadIdx.x;
  const int bhl = blockIdx.x >> 4;
  const int qt = blockIdx.x & 15;
  const size_t base = (size_t)bhl * ROWSTR;

  // stage pruned+scaled Q tile (16x32)
  At[t] = qp[base + qt * 512 + t];
  At[t + 256] = qp[base + qt * 512 + t + 256];
  __syncthreads();

  const int wave = t >> 5;
  const int lane = t & 31;
  const int hi = lane >> 4;
  const int ln16 = lane & 15;

  // ---- logits: each wave computes two 16-key tiles, K = Dh = 32 ----
  for (int kt = wave * 2; kt < wave * 2 + 2; ++kt) {
    v8f acc = {0.f, 0.f, 0.f, 0.f, 0.f, 0.f, 0.f, 0.f};
#pragma unroll
    for (int k0 = 0; k0 < 32; k0 += 4) {
      v2f a, b;
      a.x = At[ln16 * 32 + k0 + 2 * hi];
      a.y = At[ln16 * 32 + k0 + 1 + 2 * hi];
      const float* kr = kb + base + (size_t)(kt * 16 + ln16) * 32;
      b.x = kr[k0 + 2 * hi];
      b.y = kr[k0 + 1 + 2 * hi];
      acc = wmma_f32(a, b, acc);
    }
#pragma unroll
    for (int r = 0; r < 8; ++r)
      S[(r + 8 * hi) * 256 + kt * 16 + ln16] = acc[r];
  }
  __syncthreads();

  // ---- stable top-64 + softmax; 16 lanes (half-wave) per logits row ----
  const int row = t >> 4;  // lanes (row&1)*16.. of wave row>>1 -> width-16 shfl ok
  const int seg = t & 15;
  float vv[16];
  unsigned su[16];
#pragma unroll
  for (int i = 0; i < 16; ++i) {
    float val = S[row * 256 + seg * 16 + i];
    vv[i] = val;
    unsigned bts = __float_as_uint(val);
    su[i] = (bts & 0x80000000u) ? ~bts : (bts | 0x80000000u);  // order-preserving
  }
  // binary search for T = 64th-largest (as sortable uint), 32 fixed iterations
  unsigned lo = 0u, hic = 0xFFFFFFFFu;
  for (int it = 0; it < 32; ++it) {
    unsigned mid =
        (unsigned)(((unsigned long long)lo + (unsigned long long)hic + 1ull) >> 1);
    int cnt = 0;
#pragma unroll
    for (int i = 0; i < 16; ++i) cnt += (su[i] >= mid) ? 1 : 0;
    cnt += __shfl_xor(cnt, 1, 16);
    cnt += __shfl_xor(cnt, 2, 16);
    cnt += __shfl_xor(cnt, 4, 16);
    cnt += __shfl_xor(cnt, 8, 16);
    if (cnt >= TOPKK) lo = mid; else hic = mid - 1u;
  }
  const unsigned T = lo;
  // stable tie-break: keep all > T, then equals by ascending key index
  int cntG = 0, eqLoc = 0;
  int eqPref[16];
#pragma unroll
  for (int i = 0; i < 16; ++i) {
    cntG += (su[i] > T) ? 1 : 0;
    eqPref[i] = eqLoc;
    eqLoc += (su[i] == T) ? 1 : 0;
  }
  int g = cntG;
  g += __shfl_xor(g, 1, 16);
  g += __shfl_xor(g, 2, 16);
  g += __shfl_xor(g, 4, 16);
  g += __shfl_xor(g, 8, 16);
  int eqBeforeLane = 0;
  for (int s2 = 0; s2 < 16; ++s2) {
    int c = __shfl(eqLoc, s2, 16);
    if (s2 < seg) eqBeforeLane += c;
  }
  const int eqBudget = TOPKK - g;
  unsigned kept = 0u;
#pragma unroll
  for (int i = 0; i < 16; ++i) {
    bool kp = (su[i] > T) ||
              ((su[i] == T) && (eqBeforeLane + eqPref[i] < eqBudget));
    kept |= kp ? (1u << i) : 0u;
  }
  // masked softmax (max-subtracted); pruned queries -> uniform 1/64 over keys 0..63
  float pm = -3.402823466e38f;
#pragma unroll
  for (int i = 0; i < 16; ++i)
    if (kept & (1u << i)) pm = fmaxf(pm, vv[i]);
  pm = fmaxf(pm, __shfl_xor(pm, 1, 16));
  pm = fmaxf(pm, __shfl_xor(pm, 2, 16));
  pm = fmaxf(pm, __shfl_xor(pm, 4, 16));
  pm = fmaxf(pm, __shfl_xor(pm, 8, 16));
  float ps = 0.f;
#pragma unroll
  for (int i = 0; i < 16; ++i) {
    float a = (kept & (1u << i)) ? expf(vv[i] - pm) : 0.f;
    vv[i] = a;
    ps += a;
  }
  ps += __shfl_xor(ps, 1, 16);
  ps += __shfl_xor(ps, 2, 16);
  ps += __shfl_xor(ps, 4, 16);
  ps += __shfl_xor(ps, 8, 16);
  const float inv = 1.f / ps;
#pragma unroll
  for (int i = 0; i < 16; ++i) S[row * 256 + seg * 16 + i] = vv[i] * inv;
  __syncthreads();

  // ---- att(16x256) @ V(256x32): waves 0,1 each do one 16-col tile ----
  if (wave < 2) {
    const int ntile = wave;
    v8f acc = {0.f, 0.f, 0.f, 0.f, 0.f, 0.f, 0.f, 0.f};
#pragma unroll 4
    for (int k0 = 0; k0 < 256; k0 += 4) {
      v2f a, b;
      a.x = S[ln16 * 256 + k0 + 2 * hi];
      a.y = S[ln16 * 256 + k0 + 1 + 2 * hi];
      b.x = vb[base + (size_t)(k0 + 2 * hi) * 32 + ntile * 16 + ln16];
      b.y = vb[base + (size_t)(k0 + 1 + 2 * hi) * 32 + ntile * 16 + ln16];
      acc = wmma_f32(a, b, acc);
    }
#pragma unroll
    for (int r = 0; r < 8; ++r) {
      int M = r + 8 * hi;
      ab[base + (size_t)(qt * 16 + M) * 32 + ntile * 16 + ln16] = acc[r];
    }
  }
}

// ---------------------------------------------------------------------------
// Kernel 4: merge heads + mean over n (mean commutes with final projection)
//   abar[(b,l,w12), h*32+dh] = 0.25 * sum_n a[(b*4+h), l, n*64+w12, dh]
// ---------------------------------------------------------------------------
__global__ __launch_bounds__(256)
void merge_heads_kernel(const float* __restrict__ ab, float* __restrict__ abar) {
  const int idx = blockIdx.x * 256 + threadIdx.x;  // 16384*128 total
  const int c = idx & 127;
  const int row = idx >> 7;
  const int b = row >> 12;
  const int l = (row >> 6) & 63;
  const int w12 = row & 63;
  const int h = c >> 5, dh = c & 31;
  float s = 0.f;
#pragma unroll
  for (int n = 0; n < 4; ++n)
    s += ab[(((size_t)(b * 4 + h)) * 64 + l) * ROWSTR + (n * 64 + w12) * 32 + dh];
  abar[idx] = 0.25f * s;
}

// ---------------------------------------------------------------------------
// Kernel 5: output projection: out[16384 x 128] = abar @ w_proj + b_proj
// ---------------------------------------------------------------------------
__global__ __launch_bounds__(256)
void proj_out_kernel(const float* __restrict__ A, const float* __restrict__ W,
                     const float* __restrict__ bias, float* __restrict__ out) {
  __shared__ float As[16 * 128];
  const int t = threadIdx.x;
  const int rowbase = blockIdx.x * 16;
#pragma unroll
  for (int i = 0; i < 8; ++i) {
    int idx = t + 256 * i;
    As[idx] = A[(size_t)rowbase * 128 + idx];
  }
  __syncthreads();

  const int wave = t >> 5;
  const int lane = t & 31;
  const int hi = lane >> 4;
  const int ln16 = lane & 15;
  const int tileN = wave * 16;

  v8f acc = {0.f, 0.f, 0.f, 0.f, 0.f, 0.f, 0.f, 0.f};
#pragma unroll 4
  for (int k0 = 0; k0 < 128; k0 += 4) {
    v2f a, b;
    a.x = As[ln16 * 128 + k0 + 2 * hi];
    a.y = As[ln16 * 128 + k0 + 1 + 2 * hi];
    b.x = W[(size_t)(k0 + 2 * hi) * 128 + tileN + ln16];
    b.y = W[(size_t)(k0 + 1 + 2 * hi) * 128 + tileN + ln16];
    acc = wmma_f32(a, b, acc);
  }
  const int col = tileN + ln16;
  const float bv = bias[col];
#pragma unroll
  for (int r = 0; r < 8; ++r) {
    int M = r + 8 * hi;
    out[(size_t)(rowbase + M) * 128 + col] = acc[r] + bv;
  }
}

// ---------------------------------------------------------------------------
extern "C" void kernel_launch(void* const* d_in, const int* in_sizes, int n_in,
                              void* d_out, int out_size, void* d_ws,
                              size_t ws_size, hipStream_t stream) {
  (void)in_sizes; (void)n_in; (void)out_size; (void)ws_size;
  const float* q      = (const float*)d_in[0];
  const float* k      = (const float*)d_in[1];
  const float* v      = (const float*)d_in[2];
  const float* ln_q_g = (const float*)d_in[3];
  const float* ln_q_b = (const float*)d_in[4];
  const float* ln_k_g = (const float*)d_in[5];
  const float* ln_k_b = (const float*)d_in[6];
  const float* ln_v_g = (const float*)d_in[7];
  const float* ln_v_b = (const float*)d_in[8];
  const float* wq     = (const float*)d_in[9];
  const float* bq     = (const float*)d_in[10];
  const float* wk     = (const float*)d_in[11];
  const float* bk     = (const float*)d_in[12];
  const float* wv     = (const float*)d_in[13];
  const float* bv     = (const float*)d_in[14];
  const float* w_proj = (const float*)d_in[15];
  const float* b_proj = (const float*)d_in[16];
  float* out = (float*)d_out;

  float* ws = (float*)d_ws;
  const size_t SLAB = (size_t)BHTOT * LL * QQ * DHEAD;  // 8,388,608 floats
  float* qbuf = ws;
  float* kbuf = ws + SLAB;
  float* vbuf = ws + 2 * SLAB;
  float* abuf = ws + 3 * SLAB;
  float* abar = qbuf;  // q buffer is dead after attention; overlay the average

  // 1) LN + QKV projections: 65536 rows -> 4096 blocks of 16 rows each
  ln_proj_kernel<<<4096, 256, 0, stream>>>(q, ln_q_g, ln_q_b, wq, bq, qbuf);
  ln_proj_kernel<<<4096, 256, 0, stream>>>(k, ln_k_g, ln_k_b, wk, bk, kbuf);
  ln_proj_kernel<<<4096, 256, 0, stream>>>(v, ln_v_g, ln_v_b, wv, bv, vbuf);
  // 2) query saliency pruning (top-192) + 1/sqrt(Dh) pre-scale
  prune_q_kernel<<<BHTOT * LL, 256, 0, stream>>>(qbuf);
  // 3) attention: one block per (bh, l, 16-query tile)
  attn_kernel<<<BHTOT * LL * 16, 256, 0, stream>>>(qbuf, kbuf, vbuf, abuf);
  // 4) merge heads + mean over n
  merge_heads_kernel<<<(16384 * 128) / 256, 256, 0, stream>>>(abuf, abar);
  // 5) output projection straight into d_out
  proj_out_kernel<<<16384 / 16, 256, 0, stream>>>(abar, w_proj, b_proj, out);
}